// EdgeAwareLoss_1606317769444
// MI455X (gfx1250) — compile-verified
//
#include <hip/hip_runtime.h>
#include <hip/hip_bf16.h>

// Shapes fixed by the reference harness.
#define BB 8
#define CC 20
#define HH 512
#define WW 512
#define HW (HH * WW)
#define NPIX (BB * HW)
#define TPB 256
#define NBLK (NPIX / TPB)   // 8192, multiple of 64

typedef __attribute__((ext_vector_type(2))) float v2f;
typedef __attribute__((ext_vector_type(8))) float v8f;

__device__ __forceinline__ int map_label(int x) { return (x == 255) ? -1 : x; }

// Register-resident masked NLL for one pixel: one pass, one exp per element.
__device__ __forceinline__ float nll20(const float* __restrict__ base, int hw, int tsafe) {
    float xs[CC];
#pragma unroll
    for (int c = 0; c < CC; ++c)
        xs[c] = base[(size_t)c * HW + hw];
    float m = xs[0];
#pragma unroll
    for (int c = 1; c < CC; ++c)
        m = fmaxf(m, xs[c]);
    float s = 0.0f, xt = xs[0];
#pragma unroll
    for (int c = 0; c < CC; ++c) {
        s += __expf(xs[c] - m);
        xt = (c == tsafe) ? xs[c] : xt;
    }
    return (m + __logf(s)) - xt;   // -log_softmax[target]
}

__global__ void __launch_bounds__(TPB)
edge_loss_partials(const float* __restrict__ coarse,
                   const float* __restrict__ refined,
                   const float* __restrict__ edgel,
                   const int*   __restrict__ targets,
                   float*       __restrict__ part) {
    const int p  = blockIdx.x * TPB + threadIdx.x;       // grid covers NPIX exactly
    const int b  = p / HW;
    const int hw = p - b * HW;
    const int h  = hw / WW;
    const int w  = hw - h * WW;

    const int t = targets[p];
    const bool valid = (t != 255);
    const float vf = valid ? 1.0f : 0.0f;

    // --- edge target: np.gradient nonzero iff operand labels differ ---
    const int hp = (h == 0) ? 0 : h - 1;
    const int hn = (h == HH - 1) ? (HH - 1) : h + 1;
    const int wp = (w == 0) ? 0 : w - 1;
    const int wn = (w == WW - 1) ? (WW - 1) : w + 1;
    const int* tb = targets + (size_t)b * HW;
    const int thp = tb[hp * WW + w];
    const int thn = tb[hn * WW + w];
    const int twp = tb[h * WW + wp];
    const int twn = tb[h * WW + wn];
    const bool eg = (map_label(thp) != map_label(thn)) ||
                    (map_label(twp) != map_label(twn));

    // --- BCE-with-logits term (masked z, masked y; mean over ALL pixels) ---
    const float y = (eg ? 1.0f : 0.0f) * vf;
    const float z = edgel[p] * vf;
    const float bce = fmaxf(z, 0.0f) - z * y + log1pf(__expf(-fabsf(z)));

    // --- masked NLL for both logit tensors ---
    const int tsafe = valid ? t : 0;
    const size_t pixbase = (size_t)b * CC * HW;
    float a0 = nll20(coarse  + pixbase, hw, tsafe) * vf;
    float a1 = nll20(refined + pixbase, hw, tsafe) * vf;
    float a2 = vf;
    float a3 = bce;

    // --- wave32 reduction (deterministic) ---
#pragma unroll
    for (int off = 16; off > 0; off >>= 1) {
        a0 += __shfl_xor(a0, off, 32);
        a1 += __shfl_xor(a1, off, 32);
        a2 += __shfl_xor(a2, off, 32);
        a3 += __shfl_xor(a3, off, 32);
    }

    __shared__ float4 wsum[TPB / 32];
    const int lane = threadIdx.x & 31;
    const int wid  = threadIdx.x >> 5;
    if (lane == 0) wsum[wid] = make_float4(a0, a1, a2, a3);
    __syncthreads();
    if (threadIdx.x == 0) {
        float4 r = wsum[0];
#pragma unroll
        for (int i = 1; i < TPB / 32; ++i) {
            r.x += wsum[i].x; r.y += wsum[i].y;
            r.z += wsum[i].z; r.w += wsum[i].w;
        }
        part[0 * NBLK + blockIdx.x] = r.x;   // sum nll_coarse * valid
        part[1 * NBLK + blockIdx.x] = r.y;   // sum nll_refined * valid
        part[2 * NBLK + blockIdx.x] = r.z;   // sum valid
        part[3 * NBLK + blockIdx.x] = r.w;   // sum bce
    }
}

// Finalize: each of 4 waves reduces one 8192-float partial array with an
// accumulating chain of V_WMMA_F32_16X16X4_F32 (D = A x ones + C): one WMMA
// folds 64 lane-held floats into the 16 f32 row-sums of the accumulator.
__global__ void __launch_bounds__(128)
edge_loss_finalize(const float* __restrict__ part, float* __restrict__ out) {
    __shared__ float totals[4];
    const int lane = threadIdx.x & 31;
    const int wid  = threadIdx.x >> 5;

    if (wid < 4) {                       // whole-wave branch: EXEC stays all-ones
        const float* q = part + (size_t)wid * NBLK;
        v8f acc = {};                    // C/D accumulator (16x16 f32, cols identical)
        v2f ones; ones[0] = 1.0f; ones[1] = 1.0f;   // B = 4x16 all-ones
        for (int i = 0; i < NBLK; i += 64) {
            v2f a;
            a[0] = q[i + lane];
            a[1] = q[i + 32 + lane];
            acc = __builtin_amdgcn_wmma_f32_16x16x4_f32(
                false, a, false, ones, (short)0, acc, false, false);
        }
        // Every column of D equals the row-sum vector: lanes 0-15 hold rows 0-7
        // in acc[0..7], lanes 16-31 hold rows 8-15. Fold rows, then halves.
        float su = acc[0] + acc[1] + acc[2] + acc[3] +
                   acc[4] + acc[5] + acc[6] + acc[7];
        su += __shfl_xor(su, 16, 32);
        if (lane == 0) totals[wid] = su;
    }
    __syncthreads();
    if (threadIdx.x == 0) {
        const float vs = fmaxf(totals[2], 1.0f);
        out[0] = totals[0] / vs + totals[1] / vs
               + 0.4f * (totals[3] / (float)NPIX);
    }
}

extern "C" void kernel_launch(void* const* d_in, const int* in_sizes, int n_in,
                              void* d_out, int out_size, void* d_ws, size_t ws_size,
                              hipStream_t stream) {
    const float* coarse  = (const float*)d_in[0];
    const float* refined = (const float*)d_in[1];
    const float* edgel   = (const float*)d_in[2];
    const int*   targets = (const int*)d_in[3];
    float* part = (float*)d_ws;          // 4 * NBLK floats = 128 KB
    float* out  = (float*)d_out;

    edge_loss_partials<<<NBLK, TPB, 0, stream>>>(coarse, refined, edgel, targets, part);
    edge_loss_finalize<<<1, 128, 0, stream>>>(part, out);
}